// EmbEdgeGNNGRU_11141145166541
// MI455X (gfx1250) — compile-verified
//
#include <hip/hip_runtime.h>
#include <hip/hip_bf16.h>
#include <math.h>
#include <stdint.h>

// Problem constants (fixed by the reference)
#define NN   65536   // nodes
#define BB   1024    // graphs
#define NP   64      // nodes per graph
#define EP   512     // edges per graph
#define EE   (BB*EP) // 524288 edges
#define HID  128
#define DIN  64

typedef __attribute__((ext_vector_type(2))) float v2f;
typedef __attribute__((ext_vector_type(8))) float v8f;

__device__ __forceinline__ float gelu_exact(float x) {
    return 0.5f * x * (1.0f + erff(x * 0.70710678118654752440f));
}

// gfx1250 async global->LDS copy (ASYNCcnt-tracked), 16 bytes per lane.
// vdst operand = LDS byte offset (low 32 bits of the generic shared-pointer).
__device__ __forceinline__ void async_g2lds_b128(void* lds_ptr, const void* gptr) {
    uint32_t lds_off = (uint32_t)(uintptr_t)lds_ptr;
    asm volatile("global_load_async_to_lds_b128 %0, %1, off"
                 :: "v"(lds_off), "v"(gptr)
                 : "memory");
}
__device__ __forceinline__ void wait_async_zero() {
    asm volatile("s_wait_asynccnt 0x0" ::: "memory");
}

// ---------------- degree / gcn_norm ----------------
__global__ __launch_bounds__(256) void deg_init_kernel(float* deg) {
    int i = blockIdx.x * 256 + threadIdx.x;
    if (i < NN) deg[i] = 1.0f;              // self loop
}
__global__ __launch_bounds__(256) void deg_count_kernel(const int* __restrict__ col, float* deg) {
    int e = blockIdx.x * 256 + threadIdx.x;
    if (e < EE) atomicAdd(&deg[col[e]], 1.0f);
}
__global__ __launch_bounds__(256) void deg_finalize_kernel(float* deg) {
    int i = blockIdx.x * 256 + threadIdx.x;
    if (i < NN) deg[i] = rsqrtf(deg[i]);    // deg >= 1 always
}

// ---------------- h0 = concat(feat, emb[node_id]) ----------------
__global__ __launch_bounds__(256) void build_h0_kernel(const float* __restrict__ x,
                                                       const float* __restrict__ emb,
                                                       float* __restrict__ h0) {
    int i = blockIdx.x * 256 + threadIdx.x;          // over N*128
    if (i >= NN * HID) return;
    int node = i >> 7, j = i & 127;
    float v;
    if (j < DIN) {
        v = x[node * 65 + j];
    } else {
        int id = (int)x[node * 65 + 64];
        v = emb[id * 64 + (j - DIN)];
    }
    h0[i] = v;
}

// ---------------- fp32 WMMA GEMM: C[N,NCOLS] = A[N,128] @ W[128,NCOLS] ----------------
// block = 256 threads = 8 waves; each wave computes one 16x16 tile with a K-loop of
// 32 x V_WMMA_F32_16X16X4_F32.
// W is DMA'd into LDS with global_load_async_to_lds_b128, then transposed in LDS so
// each B fragment is a single contiguous ds_load_b64 with an immediate offset.
template <int NCOLS>
__global__ __launch_bounds__(256) void gemm_wmma_f32(const float* __restrict__ A,
                                                     const float* __restrict__ W,
                                                     float* __restrict__ C) {
    __shared__ float Wraw[HID * NCOLS];              // row-major W[k][n]
    __shared__ float Wt[NCOLS * HID];                // transposed Wt[n][k]
    int tid = threadIdx.x;

    // async DMA stage: HID*NCOLS floats = HID*NCOLS/4 b128 transfers
    constexpr int NVEC = (HID * NCOLS) / 4;
    for (int i = tid; i < NVEC; i += 256)
        async_g2lds_b128(&Wraw[i * 4], W + i * 4);
    wait_async_zero();
    __syncthreads();

    // LDS transpose (compile-time NCOLS -> no div/mod at runtime cost)
    for (int i = tid; i < HID * NCOLS; i += 256) {
        int k = i / NCOLS, n = i % NCOLS;
        Wt[n * HID + k] = Wraw[i];
    }
    __syncthreads();

    int wave = tid >> 5, lane = tid & 31;
    constexpr int wavesPerRow = NCOLS / 16;          // 8 (NCOLS=128) or 4 (NCOLS=64)
    constexpr int rowsPerBlock = 8 / wavesPerRow;    // 1 or 2
    int rowTile = blockIdx.x * rowsPerBlock + (wave / wavesPerRow);
    int colTile = wave % wavesPerRow;
    int r0 = rowTile * 16;
    int c0 = colTile * 16;

    int half = lane >> 4;                            // K sub-block select
    int l    = lane & 15;                            // M (for A) / N (for B)
    const float* Arow  = A + (size_t)(r0 + l) * HID; // A fragment rows (global, b64 loads)
    const float* Bcol  = &Wt[(c0 + l) * HID];        // B fragment column (LDS, b64 loads)

    v8f acc = {};
    #pragma unroll
    for (int k = 0; k < HID; k += 4) {
        int kk = k + 2 * half;                       // always even -> 8B aligned
        v2f a = *(const v2f*)(Arow + kk);
        v2f b = *(const v2f*)(Bcol + kk);
        acc = __builtin_amdgcn_wmma_f32_16x16x4_f32(
            /*neg_a=*/false, a, /*neg_b=*/false, b,
            /*c_mod=*/(short)0, acc, /*reuse_a=*/false, /*reuse_b=*/false);
    }

    // C/D layout: VGPR v -> lanes 0-15: M=v, N=l ; lanes 16-31: M=v+8, N=l
    float* Crow = C + c0 + l;
    #pragma unroll
    for (int v = 0; v < 8; ++v)
        Crow[(size_t)(r0 + v + 8 * half) * NCOLS] = acc[v];
}

// ---------------- GCN aggregation: out[c] = gelu(sum_e norm*hw[row] + dinv^2*hw[c] + b) ----
// one workgroup per graph; hw tile async-DMA'd into LDS, accumulate with ds_add_f32.
__global__ __launch_bounds__(256) void gcn_agg_kernel(const float* __restrict__ hw,
                                                      const float* __restrict__ dinv,
                                                      const int* __restrict__ row,
                                                      const int* __restrict__ col,
                                                      const float* __restrict__ bias,
                                                      float* __restrict__ hout) {
    __shared__ float hwS[NP * HID];                  // 32KB
    __shared__ float accS[NP * HID];                 // 32KB
    __shared__ float dS[NP];
    int g = blockIdx.x;
    int tid = threadIdx.x;
    int nbase = g * NP;

    // async DMA stage of this graph's hw tile: 64*128 floats = 2048 b128 transfers
    const float* hwg = hw + (size_t)nbase * HID;
    for (int i = tid; i < (NP * HID) / 4; i += 256)
        async_g2lds_b128(&hwS[i * 4], hwg + i * 4);
    if (tid < NP) dS[tid] = dinv[nbase + tid];
    wait_async_zero();
    __syncthreads();

    for (int i = tid; i < NP * HID; i += 256) {
        int node = i >> 7;
        accS[i] = dS[node] * dS[node] * hwS[i];      // self-loop contribution
    }
    __syncthreads();

    int ebase = g * EP;
    // 512 edges x 32 feature-groups of 4 floats
    for (int idx = tid; idx < EP * 32; idx += 256) {
        int e  = idx >> 5;
        int fg = (idx & 31) << 2;
        int r = row[ebase + e] - nbase;
        int c = col[ebase + e] - nbase;
        float nrm = dS[r] * dS[c];
        const float* src = &hwS[r * HID + fg];
        float* dst = &accS[c * HID + fg];
        atomicAdd(&dst[0], nrm * src[0]);
        atomicAdd(&dst[1], nrm * src[1]);
        atomicAdd(&dst[2], nrm * src[2]);
        atomicAdd(&dst[3], nrm * src[3]);
    }
    __syncthreads();

    for (int i = tid; i < NP * HID; i += 256) {
        int f = i & 127;
        hout[(size_t)nbase * HID + i] = gelu_exact(accS[i] + bias[f]);
    }
}

// ---------------- FC epilogue: hfc = gelu(raw + fcb) ----------------
__global__ __launch_bounds__(256) void fc_post_kernel(const float* __restrict__ raw,
                                                      const float* __restrict__ fcb,
                                                      float* __restrict__ hfc) {
    int i = blockIdx.x * 256 + threadIdx.x;          // over N*64
    if (i >= NN * DIN) return;
    hfc[i] = gelu_exact(raw[i] + fcb[i & 63]);
}

// ---------------- per-edge GRU gate inputs: gi[e,:] = xe @ Wih^T + bih ----------------
__global__ __launch_bounds__(256) void edge_gi_kernel(const float* __restrict__ hfc,
                                                      const int* __restrict__ row,
                                                      const int* __restrict__ col,
                                                      const float* __restrict__ Wih,
                                                      const float* __restrict__ bih,
                                                      float* __restrict__ gi) {
    int e = blockIdx.x * 256 + threadIdx.x;
    if (e >= EE) return;
    const float* hr = hfc + (size_t)row[e] * DIN;
    const float* hc = hfc + (size_t)col[e] * DIN;
    float a0 = 0.f, a1 = 0.f, a2 = 0.f;
    #pragma unroll 8
    for (int j = 0; j < DIN; ++j) {
        float xe = 0.5f * (hr[j] + hc[j]);
        a0 += xe * Wih[j];
        a1 += xe * Wih[DIN + j];
        a2 += xe * Wih[2 * DIN + j];
    }
    gi[e * 3 + 0] = a0 + bih[0];
    gi[e * 3 + 1] = a1 + bih[1];
    gi[e * 3 + 2] = a2 + bih[2];
}

// ---------------- GRU scan (hidden=1), one lane per graph ----------------
__global__ __launch_bounds__(256) void gru_scan_kernel(const float* __restrict__ gi,
                                                       const float* __restrict__ Whh,
                                                       const float* __restrict__ bhh,
                                                       const float* __restrict__ init_hs,
                                                       float* __restrict__ out_sel,
                                                       float* __restrict__ out_flat) {
    int g = blockIdx.x * 256 + threadIdx.x;
    if (g >= BB) return;
    float w0 = Whh[0], w1 = Whh[1], w2 = Whh[2];
    float c0 = bhh[0], c1 = bhh[1], c2 = bhh[2];
    float h = init_hs[0];
    float sel = 0.0f;
    const float* gp = gi + (size_t)g * EP * 3;
    float* fp = out_flat + (size_t)g * EP;
    for (int t = 0; t < EP; ++t) {
        float g0 = gp[t * 3 + 0];
        float g1 = gp[t * 3 + 1];
        float g2 = gp[t * 3 + 2];
        float r = 1.0f / (1.0f + expf(-(g0 + w0 * h + c0)));
        float z = 1.0f / (1.0f + expf(-(g1 + w1 * h + c1)));
        float n = tanhf(g2 + r * (w2 * h + c2));
        h = (1.0f - z) * n + z * h;
        fp[t] = h;                                   // mask is all-ones -> masked == pred
        if (h != 0.0f) sel = h;                      // last nonzero
    }
    out_sel[g] = sel;
}

extern "C" void kernel_launch(void* const* d_in, const int* in_sizes, int n_in,
                              void* d_out, int out_size, void* d_ws, size_t ws_size,
                              hipStream_t stream) {
    const float* x    = (const float*)d_in[0];
    const float* emb  = (const float*)d_in[1];
    const float* W1   = (const float*)d_in[2];
    const float* b1   = (const float*)d_in[3];
    const float* W2   = (const float*)d_in[4];
    const float* b2   = (const float*)d_in[5];
    const float* W3   = (const float*)d_in[6];
    const float* b3   = (const float*)d_in[7];
    const float* fcW  = (const float*)d_in[8];
    const float* fcb  = (const float*)d_in[9];
    const float* Wih  = (const float*)d_in[10];
    const float* Whh  = (const float*)d_in[11];
    const float* bih  = (const float*)d_in[12];
    const float* bhh  = (const float*)d_in[13];
    const float* ihs  = (const float*)d_in[14];
    const int*   ei   = (const int*)d_in[15];
    const int*   row  = ei;
    const int*   col  = ei + EE;

    float* out = (float*)d_out;
    float* out_sel  = out;            // [1024]
    float* out_flat = out + BB;       // [524288]

    // workspace layout (floats): dinv | bufA | bufB | bufC
    float* ws   = (float*)d_ws;
    float* dinv = ws;                          // N
    float* bufA = dinv + NN;                   // N*128
    float* bufB = bufA + (size_t)NN * HID;     // N*128
    float* bufC = bufB + (size_t)NN * HID;     // N*128

    // gcn_norm
    deg_init_kernel<<<NN / 256, 256, 0, stream>>>(dinv);
    deg_count_kernel<<<EE / 256, 256, 0, stream>>>(col, dinv);
    deg_finalize_kernel<<<NN / 256, 256, 0, stream>>>(dinv);

    // h0 = concat(feat, emb[id]) -> bufA
    build_h0_kernel<<<(NN * HID) / 256, 256, 0, stream>>>(x, emb, bufA);

    // layer 1: A -> C -> B
    gemm_wmma_f32<HID><<<NN / 16, 256, 0, stream>>>(bufA, W1, bufC);
    gcn_agg_kernel<<<BB, 256, 0, stream>>>(bufC, dinv, row, col, b1, bufB);
    // layer 2: B -> C -> A
    gemm_wmma_f32<HID><<<NN / 16, 256, 0, stream>>>(bufB, W2, bufC);
    gcn_agg_kernel<<<BB, 256, 0, stream>>>(bufC, dinv, row, col, b2, bufA);
    // layer 3: A -> C -> B
    gemm_wmma_f32<HID><<<NN / 16, 256, 0, stream>>>(bufA, W3, bufC);
    gcn_agg_kernel<<<BB, 256, 0, stream>>>(bufC, dinv, row, col, b3, bufB);

    // FC: B @ fcW -> C (64 cols), gelu+bias -> A (N*64 used)
    gemm_wmma_f32<DIN><<<NN / 32, 256, 0, stream>>>(bufB, fcW, bufC);
    fc_post_kernel<<<(NN * DIN) / 256, 256, 0, stream>>>(bufC, fcb, bufA);

    // per-edge GRU gate inputs -> bufB (E*3 floats fit)
    edge_gi_kernel<<<EE / 256, 256, 0, stream>>>(bufA, row, col, Wih, bih, bufB);

    // GRU scan + outputs
    gru_scan_kernel<<<BB / 256, 256, 0, stream>>>(bufB, Whh, bhh, ihs, out_sel, out_flat);
}